// MeshLoss_46282567582276
// MI455X (gfx1250) — compile-verified
//
#include <hip/hip_runtime.h>
#include <stdint.h>

// ---------------- constants from the reference ----------------
#define B_CONST   4
#define NS_CONST  4096
#define VP_CONST  2562
#define FP_CONST  5120
#define VG_CONST  6890
#define FG_CONST  13776
#define EPS_CONST 1e-12f
#define CHAMFER_W 1.0f
#define NORM_W    0.1f
#define EDGE_W    0.5f

#define CHUNK     1024   // gt points staged in LDS per block iteration (16 KB)

typedef __attribute__((ext_vector_type(2))) float v2f;
typedef __attribute__((ext_vector_type(8))) float v8f;

// ---------------- deterministic hash RNG ----------------
__device__ __forceinline__ uint32_t hash3(uint32_t a, uint32_t b, uint32_t c) {
    uint32_t h = a * 0x9E3779B1u ^ b * 0x85EBCA77u ^ c * 0xC2B2AE3Du;
    h ^= h >> 16; h *= 0x7FEB352Du;
    h ^= h >> 15; h *= 0x846CA68Bu;
    h ^= h >> 16;
    return h;
}
__device__ __forceinline__ float u01(uint32_t h) {
    return (float)(h >> 8) * (1.0f / 16777216.0f); // [0,1)
}

// ---------------- 1) face geometry: area + unit normal ----------------
__global__ void face_geom_kernel(const float* __restrict__ verts,
                                 const int* __restrict__ faces,
                                 int B, int V, int F,
                                 float* __restrict__ area,
                                 float* __restrict__ fnx,
                                 float* __restrict__ fny,
                                 float* __restrict__ fnz) {
    int idx = blockIdx.x * blockDim.x + threadIdx.x;
    if (idx >= B * F) return;
    int b = idx / F, f = idx % F;
    int i0 = faces[f * 3 + 0], i1 = faces[f * 3 + 1], i2 = faces[f * 3 + 2];
    const float* vb = verts + (size_t)b * V * 3;
    float x0 = vb[i0 * 3 + 0], y0 = vb[i0 * 3 + 1], z0 = vb[i0 * 3 + 2];
    float x1 = vb[i1 * 3 + 0], y1 = vb[i1 * 3 + 1], z1 = vb[i1 * 3 + 2];
    float x2 = vb[i2 * 3 + 0], y2 = vb[i2 * 3 + 1], z2 = vb[i2 * 3 + 2];
    float e1x = x1 - x0, e1y = y1 - y0, e1z = z1 - z0;
    float e2x = x2 - x0, e2y = y2 - y0, e2z = z2 - z0;
    float cx = e1y * e2z - e1z * e2y;
    float cy = e1z * e2x - e1x * e2z;
    float cz = e1x * e2y - e1y * e2x;
    float a2 = sqrtf(cx * cx + cy * cy + cz * cz);
    area[idx] = 0.5f * a2;
    float inv = 1.0f / (a2 + EPS_CONST);
    fnx[idx] = cx * inv; fny[idx] = cy * inv; fnz[idx] = cz * inv;
}

// ---------------- 2) per-batch inclusive scan (one block per batch) ----------------
__global__ void scan_kernel(const float* __restrict__ area,
                            float* __restrict__ cdf, int F) {
    __shared__ float buf[256];
    int b = blockIdx.x;
    float running = 0.0f;
    for (int base = 0; base < F; base += 256) {
        int i = base + (int)threadIdx.x;
        float v = (i < F) ? area[b * F + i] : 0.0f;
        buf[threadIdx.x] = v;
        __syncthreads();
        #pragma unroll
        for (int off = 1; off < 256; off <<= 1) {
            float t = (threadIdx.x >= (unsigned)off) ? buf[threadIdx.x - off] : 0.0f;
            __syncthreads();
            buf[threadIdx.x] += t;
            __syncthreads();
        }
        if (i < F) cdf[b * F + i] = buf[threadIdx.x] + running;
        running += buf[255];
        __syncthreads();
    }
}

// ---------------- 3) area-proportional sampling (packed float4 output) ----------------
__global__ void sample_kernel(const float* __restrict__ verts,
                              const int* __restrict__ faces,
                              const float* __restrict__ cdf,
                              const float* __restrict__ fnx,
                              const float* __restrict__ fny,
                              const float* __restrict__ fnz,
                              int B, int V, int F, int NS, uint32_t seed,
                              float4* __restrict__ pts4,
                              float* __restrict__ snx, float* __restrict__ sny,
                              float* __restrict__ snz) {
    int idx = blockIdx.x * blockDim.x + threadIdx.x;
    if (idx >= B * NS) return;
    int b = idx / NS;
    const float* c = cdf + (size_t)b * F;
    float total = c[F - 1];
    float r = u01(hash3(seed, (uint32_t)idx, 0u)) * total;
    int lo = 0, hi = F - 1;
    while (lo < hi) {
        int mid = (lo + hi) >> 1;
        if (c[mid] <= r) lo = mid + 1; else hi = mid;
    }
    int f = lo;
    int i0 = faces[f * 3 + 0], i1 = faces[f * 3 + 1], i2 = faces[f * 3 + 2];
    const float* vb = verts + (size_t)b * V * 3;
    float u = u01(hash3(seed, (uint32_t)idx, 1u));
    float v = u01(hash3(seed, (uint32_t)idx, 2u));
    float r1 = sqrtf(u);
    float w0 = 1.0f - r1, w1 = r1 * (1.0f - v), w2 = r1 * v;
    float px = w0 * vb[i0 * 3 + 0] + w1 * vb[i1 * 3 + 0] + w2 * vb[i2 * 3 + 0];
    float py = w0 * vb[i0 * 3 + 1] + w1 * vb[i1 * 3 + 1] + w2 * vb[i2 * 3 + 1];
    float pz = w0 * vb[i0 * 3 + 2] + w1 * vb[i1 * 3 + 2] + w2 * vb[i2 * 3 + 2];
    pts4[idx] = make_float4(px, py, pz, px * px + py * py + pz * pz);
    int fi = b * F + f;
    snx[idx] = fnx[fi]; sny[idx] = fny[fi]; snz[idx] = fnz[fi];
}

// ---------------- 4) WMMA chamfer with async global->LDS staging ----------------
// The full squared distance comes straight out of the matrix unit:
//   A = (-2x, -2y | -2z, 1)  (K=4, spare slot carries the "1")
//   B = ( gx,  gy |  gz, |g|^2)
//   C[v] = |p[row]|^2        (loop-invariant)
//   D = |p|^2 + |g|^2 - 2 p.g   -> only min/argmin VALU remains in the loop.
__global__ void __launch_bounds__(256)
chamfer_kernel(const float4* __restrict__ ppts,
               const float4* __restrict__ gpts,
               int B, int N, int M,
               float* __restrict__ minOut, int* __restrict__ argOut) {
    __shared__ float4 tile[CHUNK];

    const int lane = threadIdx.x & 31;
    const int wid = threadIdx.x >> 5;
    const int wavesPerBlock = blockDim.x >> 5;        // 8
    const int gwave = blockIdx.x * wavesPerBlock + wid;
    const int tilesPerBatch = N >> 4;                 // 256
    const int b = gwave / tilesPerBatch;              // block-uniform (8 | 256)
    const int m0 = (gwave % tilesPerBatch) << 4;
    const int half = lane >> 4;                       // 0: K={x,y}  1: K={z,1}
    const int l = lane & 15;

    // A operand: one b128 load, branchless K-pair select, pre-scaled by -2
    float4 p = ppts[b * N + m0 + l];
    v2f a;
    a.x = -2.0f * (half ? p.z : p.x);
    a.y = half ? 1.0f : (-2.0f * p.y);

    // C operand: |p|^2 broadcast per output row (D VGPR v -> row m0 + 8*half + v)
    v8f cvec;
    #pragma unroll
    for (int v = 0; v < 8; ++v) cvec[v] = __shfl(p.w, (half << 3) + v, 32);

    float best[8];
    int   barg[8];
    #pragma unroll
    for (int v = 0; v < 8; ++v) { best[v] = 3.4e38f; barg[v] = 0; }

    const float4* gb = gpts + (size_t)b * M;

    for (int c0 = 0; c0 < M; c0 += CHUNK) {
        // --- async stage: straight-line, 256 threads x CHUNK/256 float4 each ---
        {
            int i = threadIdx.x;
            #pragma unroll
            for (int r = 0; r < CHUNK / 256; ++r, i += 256) {
                uint32_t ldsAddr = (uint32_t)(uintptr_t)&tile[i];
                const float4* src = &gb[c0 + i];
                asm volatile("global_load_async_to_lds_b128 %0, %1, off"
                             :: "v"(ldsAddr), "v"(src) : "memory");
            }
        }
        asm volatile("s_wait_asynccnt 0" ::: "memory");
        __syncthreads();

        // --- sweep chunk from LDS: 64 WMMA tiles ---
        #pragma unroll 4
        for (int t = 0; t < CHUNK; t += 16) {
            float4 g = tile[t + l];                   // ds_load_b128 (broadcast)
            v2f bmat;
            bmat.x = half ? g.z : g.x;
            bmat.y = half ? g.w : g.y;                // K=3 slot carries |g|^2

            // D = |p|^2 + |g|^2 - 2 p.g  (entire distance on the matrix pipe)
            v8f d2 = __builtin_amdgcn_wmma_f32_16x16x4_f32(
                false, a, false, bmat, (short)0, cvec, false, false);

            const int n = c0 + t + l;
            #pragma unroll
            for (int v = 0; v < 8; ++v) {
                if (d2[v] < best[v]) { best[v] = d2[v]; barg[v] = n; }
            }
        }
        __syncthreads();
    }

    // min/argmin across the 16 lanes of each half (cols of the tile)
    #pragma unroll
    for (int v = 0; v < 8; ++v) {
        #pragma unroll
        for (int off = 8; off >= 1; off >>= 1) {
            float ob = __shfl_xor(best[v], off, 32);
            int   oa = __shfl_xor(barg[v], off, 32);
            if (ob < best[v]) { best[v] = ob; barg[v] = oa; }
        }
    }
    if (l == 0) {
        #pragma unroll
        for (int v = 0; v < 8; ++v) {
            int row = m0 + (half << 3) + v;
            minOut[b * N + row] = fmaxf(best[v], 0.0f);   // clamp hoisted out of loop
            argOut[b * N + row] = barg[v];
        }
    }
}

// ---------------- 5) deterministic single-block finalize ----------------
__device__ __forceinline__ float block_sum(float v, float* red) {
    red[threadIdx.x] = v;
    __syncthreads();
    #pragma unroll
    for (int off = 128; off >= 1; off >>= 1) {
        if ((int)threadIdx.x < off) red[threadIdx.x] += red[threadIdx.x + off];
        __syncthreads();
    }
    float r = red[0];
    __syncthreads();
    return r;
}

__global__ void finalize_kernel(const float* __restrict__ minP2G,
                                const float* __restrict__ minG2P,
                                const int* __restrict__ argP2G,
                                const float* __restrict__ pnx, const float* __restrict__ pny,
                                const float* __restrict__ pnz,
                                const float* __restrict__ gnx, const float* __restrict__ gny,
                                const float* __restrict__ gnz,
                                const float* __restrict__ pverts,
                                const int* __restrict__ pfaces,
                                int B, int NS, int Vp, int Fp,
                                float* __restrict__ out) {
    __shared__ float red[256];
    float s0 = 0.f, s1 = 0.f, s2 = 0.f, s3 = 0.f;
    for (int i = threadIdx.x; i < B * NS; i += 256) {
        s0 += minP2G[i];
        s1 += minG2P[i];
        int b = i / NS;
        int j = b * NS + argP2G[i];
        float cosv = fabsf(pnx[i] * gnx[j] + pny[i] * gny[j] + pnz[i] * gnz[j]);
        s2 += 1.0f - cosv;
    }
    for (int i = threadIdx.x; i < B * Fp; i += 256) {
        int b = i / Fp, f = i % Fp;
        int i0 = pfaces[f * 3 + 0], i1 = pfaces[f * 3 + 1], i2 = pfaces[f * 3 + 2];
        const float* vb = pverts + (size_t)b * Vp * 3;
        float x0 = vb[i0 * 3 + 0], y0 = vb[i0 * 3 + 1], z0 = vb[i0 * 3 + 2];
        float x1 = vb[i1 * 3 + 0], y1 = vb[i1 * 3 + 1], z1 = vb[i1 * 3 + 2];
        float x2 = vb[i2 * 3 + 0], y2 = vb[i2 * 3 + 1], z2 = vb[i2 * 3 + 2];
        float ax = x1 - x0, ay = y1 - y0, az = z1 - z0;
        float bx = x2 - x1, by = y2 - y1, bz = z2 - z1;
        float cx = x0 - x2, cy = y0 - y2, cz = z0 - z2;
        s3 += ax * ax + ay * ay + az * az
            + bx * bx + by * by + bz * bz
            + cx * cx + cy * cy + cz * cz;
    }
    s0 = block_sum(s0, red);
    s1 = block_sum(s1, red);
    s2 = block_sum(s2, red);
    s3 = block_sum(s3, red);
    if (threadIdx.x == 0) {
        float invBN = 1.0f / (float)(B * NS);
        float chamfer = s0 * invBN + s1 * invBN;
        float nloss = s2 * invBN;
        float eloss = s3 / (float)(B * Fp * 3);
        out[0] = CHAMFER_W * chamfer + NORM_W * nloss + EDGE_W * eloss;
    }
}

// ---------------- host launcher ----------------
extern "C" void kernel_launch(void* const* d_in, const int* in_sizes, int n_in,
                              void* d_out, int out_size, void* d_ws, size_t ws_size,
                              hipStream_t stream) {
    (void)in_sizes; (void)n_in; (void)out_size; (void)ws_size;
    const float* pv = (const float*)d_in[0];
    const int*   pf = (const int*)d_in[1];
    const float* gv = (const float*)d_in[2];
    const int*   gf = (const int*)d_in[3];
    float* out = (float*)d_out;

    const int B = B_CONST, NS = NS_CONST;
    const int Vp = VP_CONST, Fp = FP_CONST, Vg = VG_CONST, Fg = FG_CONST;

    float* ws = (float*)d_ws;
    size_t o = 0;
    auto take = [&](size_t n) { float* p = ws + o; o += n; return p; };

    // packed sampled points first (16B aligned: offsets are multiples of 4 floats)
    float4* pP4 = (float4*)take((size_t)B * NS * 4);
    float4* gP4 = (float4*)take((size_t)B * NS * 4);
    // sample normals (SoA)
    float* pNX = take((size_t)B * NS); float* pNY = take((size_t)B * NS);
    float* pNZ = take((size_t)B * NS);
    float* gNX = take((size_t)B * NS); float* gNY = take((size_t)B * NS);
    float* gNZ = take((size_t)B * NS);
    // face geometry + CDFs
    float* areaP = take((size_t)B * Fp); float* cdfP = take((size_t)B * Fp);
    float* fnPX = take((size_t)B * Fp); float* fnPY = take((size_t)B * Fp);
    float* fnPZ = take((size_t)B * Fp);
    float* areaG = take((size_t)B * Fg); float* cdfG = take((size_t)B * Fg);
    float* fnGX = take((size_t)B * Fg); float* fnGY = take((size_t)B * Fg);
    float* fnGZ = take((size_t)B * Fg);
    // chamfer outputs
    float* minP2G = take((size_t)B * NS);
    int*   argP2G = (int*)take((size_t)B * NS);
    float* minG2P = take((size_t)B * NS);
    int*   argG2P = (int*)take((size_t)B * NS);

    // 1) face geometry
    {
        int n = B * Fp;
        face_geom_kernel<<<(n + 255) / 256, 256, 0, stream>>>(
            pv, pf, B, Vp, Fp, areaP, fnPX, fnPY, fnPZ);
        n = B * Fg;
        face_geom_kernel<<<(n + 255) / 256, 256, 0, stream>>>(
            gv, gf, B, Vg, Fg, areaG, fnGX, fnGY, fnGZ);
    }
    // 2) CDF scans (one block per batch)
    scan_kernel<<<B, 256, 0, stream>>>(areaP, cdfP, Fp);
    scan_kernel<<<B, 256, 0, stream>>>(areaG, cdfG, Fg);
    // 3) sampling
    {
        int n = B * NS;
        sample_kernel<<<(n + 255) / 256, 256, 0, stream>>>(
            pv, pf, cdfP, fnPX, fnPY, fnPZ, B, Vp, Fp, NS, 0x1234ABCDu,
            pP4, pNX, pNY, pNZ);
        sample_kernel<<<(n + 255) / 256, 256, 0, stream>>>(
            gv, gf, cdfG, fnGX, fnGY, fnGZ, B, Vg, Fg, NS, 0x5678EF01u,
            gP4, gNX, gNY, gNZ);
    }
    // 4) chamfer both directions (B*NS/16 waves, 8 waves per block)
    {
        int waves = B * (NS >> 4);
        int blocks = waves / 8;        // 128, exact
        chamfer_kernel<<<blocks, 256, 0, stream>>>(
            pP4, gP4, B, NS, NS, minP2G, argP2G);
        chamfer_kernel<<<blocks, 256, 0, stream>>>(
            gP4, pP4, B, NS, NS, minG2P, argG2P);
    }
    // 5) finalize to scalar
    finalize_kernel<<<1, 256, 0, stream>>>(
        minP2G, minG2P, argP2G, pNX, pNY, pNZ, gNX, gNY, gNZ,
        pv, pf, B, NS, Vp, Fp, out);
}